// BiLSTM_CRF_10256381903289
// MI455X (gfx1250) — compile-verified
//
#include <hip/hip_runtime.h>
#include <stdint.h>

// Problem constants
static constexpr int Vv = 50000;
static constexpr int Kt = 32;     // tags
static constexpr int Ee = 256;    // embedding
static constexpr int Hh = 512;    // hidden
static constexpr int Bb = 64;     // batch
static constexpr int Ss = 256;    // seq len
static constexpr int H2 = 2 * Hh; // 1024

typedef __attribute__((ext_vector_type(2))) float v2f;
typedef __attribute__((ext_vector_type(8))) float v8f;

// D = A(16x4 f32) x B(4x16 f32) + C(16x16 f32), exact fp32 matrix op on CDNA5
static __device__ inline v8f wmma4(v2f a, v2f b, v8f c) {
    return __builtin_amdgcn_wmma_f32_16x16x4_f32(
        /*neg_a=*/false, a, /*neg_b=*/false, b,
        /*c_mod=*/(short)0, c, /*reuse_a=*/false, /*reuse_b=*/false);
}

static __device__ inline float wave_sum(float v) {
    #pragma unroll
    for (int o = 16; o > 0; o >>= 1) v += __shfl_down(v, o, 32);
    return v; // valid on lane 0
}

__global__ void zero_kernel(float* __restrict__ p, int n) {
    int i = blockIdx.x * blockDim.x + threadIdx.x;
    if (i < n) p[i] = 0.f;
}

// One timestep of both LSTM directions, fused input GEMM (emb gather) +
// recurrent GEMM + cell. Block = 4 waves (one per gate), tile = 16 batch x 16 hidden.
__global__ __launch_bounds__(128)
void lstm_step_kernel(const int* __restrict__ words,
                      const float* __restrict__ emb,
                      const float* __restrict__ w_ih_f, const float* __restrict__ w_hh_f,
                      const float* __restrict__ b_f,
                      const float* __restrict__ w_ih_b, const float* __restrict__ w_hh_b,
                      const float* __restrict__ b_b,
                      float* __restrict__ h_state,  // [2 dirs][2 bufs][B][H]
                      float* __restrict__ c_state,  // [2 dirs][B][H]
                      float* __restrict__ hbuf,     // [B][S][2H]
                      int t)
{
    const int dir  = blockIdx.z;
    const int m0   = blockIdx.y * 16;   // batch tile
    const int j0   = blockIdx.x * 16;   // hidden tile
    const int lane = threadIdx.x & 31;
    const int wave = threadIdx.x >> 5;  // gate index: 0=i 1=f 2=g 3=o

    const float* w_ih = dir ? w_ih_b : w_ih_f;
    const float* w_hh = dir ? w_hh_b : w_hh_f;
    const float* bias = dir ? b_b    : b_f;
    const int t_src = dir ? (Ss - 1 - t) : t;

    const float* h_read  = h_state + (size_t)(dir * 2 + (t & 1)) * Bb * Hh;
    float*       h_write = h_state + (size_t)(dir * 2 + ((t + 1) & 1)) * Bb * Hh;
    float*       c_st    = c_state + (size_t)dir * Bb * Hh;

    __shared__ float gt[4][16][17];

    // ---- WMMA GEMM: gate tile ----
    const int mrow  = lane & 15;          // A row / B col within tile
    const int khalf = (lane >> 4) * 2;    // K sub-offset per lane half
    const int bi    = m0 + mrow;          // batch row (A)
    const int ncol  = wave * Hh + j0 + mrow; // gate-weight row (B source)

    v8f acc = {0.f, 0.f, 0.f, 0.f, 0.f, 0.f, 0.f, 0.f};

    {   // recurrent part: K = H over h_prev, W_hh row-major (4H, H)
        const float* ar = h_read + (size_t)bi * Hh + khalf;
        const float* br = w_hh   + (size_t)ncol * Hh + khalf;
        #pragma unroll 8
        for (int k = 0; k < Hh; k += 4) {
            v2f a = *(const v2f*)(ar + k);
            v2f b = *(const v2f*)(br + k);
            acc = wmma4(a, b, acc);
        }
    }
    {   // input part: K = E over gathered embedding row, W_ih row-major (4H, E)
        const int w = words[bi * Ss + t_src];
        const float* ar = emb  + (size_t)w * Ee + khalf;
        const float* br = w_ih + (size_t)ncol * Ee + khalf;
        #pragma unroll 8
        for (int k = 0; k < Ee; k += 4) {
            v2f a = *(const v2f*)(ar + k);
            v2f b = *(const v2f*)(br + k);
            acc = wmma4(a, b, acc);
        }
    }
    const float bv = bias[ncol];
    #pragma unroll
    for (int r = 0; r < 8; ++r)
        gt[wave][r + (lane >> 4) * 8][mrow] = acc[r] + bv;

    __syncthreads();

    // ---- LSTM cell, 256 elements over 128 threads ----
    #pragma unroll
    for (int e = threadIdx.x; e < 256; e += 128) {
        const int m = e >> 4, j = e & 15;
        const float ig = 1.f / (1.f + __expf(-gt[0][m][j]));
        const float fg = 1.f / (1.f + __expf(-gt[1][m][j]));
        const float gg = tanhf(gt[2][m][j]);
        const float og = 1.f / (1.f + __expf(-gt[3][m][j]));
        const int bi2 = m0 + m;
        const int jg  = j0 + j;
        const size_t ci = (size_t)bi2 * Hh + jg;
        const float cn = fg * c_st[ci] + ig * gg;
        c_st[ci] = cn;
        const float hn = og * tanhf(cn);
        h_write[ci] = hn;
        const int t_out = dir ? (Ss - 1 - t) : t;
        hbuf[((size_t)bi2 * Ss + t_out) * H2 + dir * Hh + jg] = hn;
    }
}

// logits = hbuf(16384 x 1024) @ w_out(1024 x 32) + b_out, WMMA fp32
__global__ __launch_bounds__(32)
void logits_kernel(const float* __restrict__ hbuf,
                   const float* __restrict__ w_out, const float* __restrict__ b_out,
                   float* __restrict__ logits)
{
    const int m0   = blockIdx.y * 16;
    const int n0   = blockIdx.x * 16;
    const int lane = threadIdx.x & 31;
    const int mrow  = lane & 15;
    const int khalf = (lane >> 4) * 2;
    const int ncol  = n0 + mrow;

    v8f acc = {0.f, 0.f, 0.f, 0.f, 0.f, 0.f, 0.f, 0.f};
    const float* ar = hbuf + (size_t)(m0 + mrow) * H2 + khalf;
    #pragma unroll 8
    for (int k = 0; k < H2; k += 4) {
        v2f a = *(const v2f*)(ar + k);
        v2f b;
        b.x = w_out[(size_t)(k + khalf) * Kt + ncol];
        b.y = w_out[(size_t)(k + khalf + 1) * Kt + ncol];
        acc = wmma4(a, b, acc);
    }
    const float bv = b_out[ncol];
    #pragma unroll
    for (int r = 0; r < 8; ++r) {
        const int m = m0 + r + (lane >> 4) * 8;
        logits[(size_t)m * Kt + ncol] = acc[r] + bv;
    }
}

// CRF numerator: one wave per sequence
__global__ __launch_bounds__(32)
void crf_numer_kernel(const float* __restrict__ logits,
                      const int* __restrict__ tags,
                      const uint8_t* __restrict__ mask,
                      const float* __restrict__ start_trans,
                      const float* __restrict__ trans,
                      const float* __restrict__ end_trans,
                      float* __restrict__ numer)
{
    const int b = blockIdx.x;
    const int lane = threadIdx.x;
    float acc = 0.f, cnt = 0.f;
    for (int t = lane; t < Ss; t += 32) {
        const float mk = mask[b * Ss + t] ? 1.f : 0.f;
        cnt += mk;
        if (t >= 1) {
            const int tg = tags[b * Ss + t];
            const int tp = tags[b * Ss + t - 1];
            acc += (logits[((size_t)b * Ss + t) * Kt + tg] + trans[tp * Kt + tg]) * mk;
        }
    }
    acc = wave_sum(acc);
    cnt = wave_sum(cnt);
    if (lane == 0) {
        const int t0 = tags[b * Ss];
        float num = start_trans[t0] + logits[((size_t)b * Ss) * Kt + t0] + acc;
        const int se = (int)(cnt + 0.5f) - 1;
        num += end_trans[tags[b * Ss + se]];
        numer[b] = num;
    }
}

// CRF forward algorithm: one wave per sequence, lane = tag, online logsumexp
__global__ __launch_bounds__(32)
void crf_fwd_kernel(const float* __restrict__ logits,
                    const uint8_t* __restrict__ mask,
                    const float* __restrict__ start_trans,
                    const float* __restrict__ trans,
                    const float* __restrict__ end_trans,
                    const float* __restrict__ numer,
                    float* __restrict__ llh)
{
    const int b = blockIdx.x;
    const int lane = threadIdx.x;

    float trc[Kt]; // column `lane` of trans, cached in registers
    #pragma unroll
    for (int k = 0; k < Kt; ++k) trc[k] = trans[k * Kt + lane];

    float alpha = start_trans[lane] + logits[((size_t)b * Ss) * Kt + lane];
    for (int t = 1; t < Ss; ++t) {
        const float em = logits[((size_t)b * Ss + t) * Kt + lane];
        float m = -3.4e38f, s = 0.f;
        #pragma unroll
        for (int k = 0; k < Kt; ++k) {
            const float v = __shfl(alpha, k, 32) + trc[k];
            if (v > m) { s = s * __expf(m - v) + 1.f; m = v; }
            else       { s += __expf(v - m); }
        }
        const float nxt = em + m + __logf(s);
        alpha = mask[b * Ss + t] ? nxt : alpha;
    }
    // log_z = logsumexp(alpha + end_trans)
    const float v = alpha + end_trans[lane];
    float mx = v;
    #pragma unroll
    for (int o = 16; o > 0; o >>= 1) mx = fmaxf(mx, __shfl_xor(mx, o, 32));
    float se = __expf(v - mx);
    #pragma unroll
    for (int o = 16; o > 0; o >>= 1) se += __shfl_xor(se, o, 32);
    if (lane == 0) llh[b] = numer[b] - (mx + __logf(se));
}

__global__ __launch_bounds__(32)
void reduce_kernel(const float* __restrict__ llh, float* __restrict__ out)
{
    const int lane = threadIdx.x;
    float v = llh[lane] + llh[lane + 32];
    #pragma unroll
    for (int o = 16; o > 0; o >>= 1) v += __shfl_down(v, o, 32);
    if (lane == 0) out[0] = -(v / (float)Bb);
}

extern "C" void kernel_launch(void* const* d_in, const int* in_sizes, int n_in,
                              void* d_out, int out_size, void* d_ws, size_t ws_size,
                              hipStream_t stream)
{
    (void)in_sizes; (void)n_in; (void)out_size; (void)ws_size;

    const int*     words  = (const int*)d_in[0];
    const int*     tags   = (const int*)d_in[1];
    const uint8_t* mask   = (const uint8_t*)d_in[2];
    const float*   emb    = (const float*)d_in[3];
    const float*   w_ih_f = (const float*)d_in[4];
    const float*   w_hh_f = (const float*)d_in[5];
    const float*   b_f    = (const float*)d_in[6];
    const float*   w_ih_b = (const float*)d_in[7];
    const float*   w_hh_b = (const float*)d_in[8];
    const float*   b_b    = (const float*)d_in[9];
    const float*   w_out  = (const float*)d_in[10];
    const float*   b_out  = (const float*)d_in[11];
    const float*   start_trans = (const float*)d_in[12];
    const float*   trans  = (const float*)d_in[13];
    const float*   end_trans   = (const float*)d_in[14];

    // Workspace layout (floats):
    float* ws      = (float*)d_ws;
    float* hbuf    = ws;                               // B*S*2H  = 16.8M floats
    float* logits  = hbuf   + (size_t)Bb * Ss * H2;    // B*S*K   = 524288
    float* h_state = logits + (size_t)Bb * Ss * Kt;    // 4*B*H (2 dirs x 2 bufs)
    float* c_state = h_state + (size_t)4 * Bb * Hh;    // 2*B*H
    float* numer   = c_state + (size_t)2 * Bb * Hh;    // B
    float* llh     = numer + Bb;                       // B

    // Zero LSTM h/c state (contiguous 6*B*H floats starting at h_state)
    {
        const int n = 6 * Bb * Hh;
        zero_kernel<<<(n + 255) / 256, 256, 0, stream>>>(h_state, n);
    }

    // Sequential scan: 256 stream-ordered step kernels; both directions via grid.z
    for (int t = 0; t < Ss; ++t) {
        lstm_step_kernel<<<dim3(Hh / 16, Bb / 16, 2), 128, 0, stream>>>(
            words, emb, w_ih_f, w_hh_f, b_f, w_ih_b, w_hh_b, b_b,
            h_state, c_state, hbuf, t);
    }

    logits_kernel<<<dim3(Kt / 16, (Bb * Ss) / 16), 32, 0, stream>>>(hbuf, w_out, b_out, logits);
    crf_numer_kernel<<<Bb, 32, 0, stream>>>(logits, tags, mask, start_trans, trans, end_trans, numer);
    crf_fwd_kernel<<<Bb, 32, 0, stream>>>(logits, mask, start_trans, trans, end_trans, numer, llh);
    reduce_kernel<<<1, 32, 0, stream>>>(llh, (float*)d_out);
}